// DiffSortNet_68023692034618
// MI455X (gfx1250) — compile-verified
//
#include <hip/hip_runtime.h>
#include <math.h>

// DiffSort (odd-even transposition soft sort), SIZE=128, BATCH=64.
// X = I * M_1 * ... * M_128  (per-batch column-mix matrices).
// Strategy: scan for alphas -> build A = M_1..M_64 and B = M_65..M_128 as
// dense matrices in LDS via column ops -> X = A @ B with v_wmma_f32_16x16x4_f32.

typedef __attribute__((ext_vector_type(2))) float v2f;
typedef __attribute__((ext_vector_type(8))) float v8f;

#define SIZE   128
#define BATCH  64
#define NPAIR  64
#define NLAYER 128
#define LDA    130   // padded LDS row stride in floats: even (b64 align), 130%64==2 spreads banks

__global__ __launch_bounds__(256)
void diffsort_fused_kernel(const float* __restrict__ vin, float* __restrict__ out) {
  extern __shared__ float lds[];
  float* Am     = lds;                      // 128 x LDA, chunk A, row-major: Am[r*LDA + c]
  float* Bt     = Am + SIZE * LDA;          // 128 x LDA, chunk B TRANSPOSED: Bt[c*LDA + k] = B[k][c]
  float* alphas = Bt + SIZE * LDA;          // 128 layers x 64 pairs
  float* xv     = alphas + NLAYER * NPAIR;  // 128 evolving x values

  const int t     = threadIdx.x;
  const int batch = blockIdx.x;
  const float inv_pi = 0.3183098861837907f;

  // ---------- Phase 1: sequential scan, record alphas, emit x_final ----------
  if (t < SIZE) xv[t] = vin[batch * SIZE + t];
  __syncthreads();

  for (int l = 0; l < NLAYER; ++l) {
    const int start = l & 1;
    const int np = (SIZE - start) >> 1;     // 64 (even layers) / 63 (odd layers)
    if (t < np) {                           // each thread owns one disjoint pair
      const int a = start + 2 * t;
      const float xa = xv[a], xb = xv[a + 1];
      const float alpha = atanf(10.0f * (xb - xa)) * inv_pi + 0.5f;
      alphas[l * NPAIR + t] = alpha;
      xv[a]     = alpha * xa + (1.0f - alpha) * xb;
      xv[a + 1] = (1.0f - alpha) * xa + alpha * xb;
    }
    __syncthreads();
  }
  if (t < SIZE) out[batch * SIZE + t] = xv[t];

  // ---------- Phase 2: build chunk matrices in LDS, starting from identity ----------
  for (int idx = t; idx < SIZE * SIZE; idx += 256) {
    const int r = idx >> 7, c = idx & 127;
    const float v = (r == c) ? 1.0f : 0.0f;
    Am[r * LDA + c] = v;
    Bt[r * LDA + c] = v;
  }
  __syncthreads();

  // Chunk A = M_1..M_64: column ops on row-major Am (consecutive threads -> consecutive rows)
  for (int l = 0; l < 64; ++l) {
    const int start = l & 1;
    const int np = (SIZE - start) >> 1;
    for (int u = t; u < np * SIZE; u += 256) {
      const int p = u >> 7, r = u & 127;
      const int a = start + 2 * p;
      const float w = alphas[l * NPAIR + p];
      const float va = Am[r * LDA + a], vb = Am[r * LDA + a + 1];
      Am[r * LDA + a]     = w * va + (1.0f - w) * vb;
      Am[r * LDA + a + 1] = (1.0f - w) * va + w * vb;
    }
    __syncthreads();
  }

  // Chunk B = M_65..M_128: column ops on B == row ops on Bt (contiguous, conflict-free)
  for (int l = 64; l < 128; ++l) {
    const int start = l & 1;
    const int np = (SIZE - start) >> 1;
    for (int u = t; u < np * SIZE; u += 256) {
      const int p = u >> 7, j = u & 127;
      const int a = start + 2 * p;
      const float w = alphas[l * NPAIR + p];
      const float va = Bt[a * LDA + j], vb = Bt[(a + 1) * LDA + j];
      Bt[a * LDA + j]       = w * va + (1.0f - w) * vb;
      Bt[(a + 1) * LDA + j] = (1.0f - w) * va + w * vb;
    }
    __syncthreads();
  }

  // ---------- Phase 3: X = A @ B via v_wmma_f32_16x16x4_f32 ----------
  // 8 waves; wave w owns output tile-row mi=w (rows 16w..16w+15), all 8 tile-cols.
  // A-frag layout (16x4 f32): lanes 0-15 -> M=lane, K={k,k+1}; lanes 16-31 -> M=lane-16, K={k+2,k+3}.
  // B-frag mirrors with N across lanes; Bt storage makes B[k][c],B[k+1][c] a contiguous b64.
  const int wave  = t >> 5;
  const int lane  = t & 31;
  const int l15   = lane & 15;
  const int lhalf = lane >> 4;         // 0: K pair {0,1}; 1: K pair {2,3}
  const int kSel  = lhalf * 2;
  const int mi    = wave;

  // Hoist all A fragments for this tile-row (32 k-steps x float2 = 64 VGPRs)
  v2f afrag[32];
  const int arow = mi * 16 + l15;
  #pragma unroll
  for (int ks = 0; ks < 32; ++ks) {
    afrag[ks] = *(const v2f*)(Am + arow * LDA + 4 * ks + kSel);   // ds_load_b64, 8B-aligned
  }

  float* outX = out + (BATCH * SIZE) + (size_t)batch * SIZE * SIZE;

  for (int ni = 0; ni < 8; ++ni) {
    v8f acc = {};
    const int bcol = ni * 16 + l15;
    #pragma unroll
    for (int ks = 0; ks < 32; ++ks) {
      const v2f bfrag = *(const v2f*)(Bt + bcol * LDA + 4 * ks + kSel);
      acc = __builtin_amdgcn_wmma_f32_16x16x4_f32(
          /*neg_a=*/false, afrag[ks], /*neg_b=*/false, bfrag,
          /*c_mod=*/(short)0, acc, /*reuse_a=*/false, /*reuse_b=*/false);
    }
    // 16x16 f32 C/D layout: VGPR v -> M = v + 8*lhalf, N = l15
    #pragma unroll
    for (int v = 0; v < 8; ++v) {
      const int row = mi * 16 + v + lhalf * 8;
      const int col = ni * 16 + l15;
      outX[row * SIZE + col] = acc[v];
    }
  }
}

extern "C" void kernel_launch(void* const* d_in, const int* in_sizes, int n_in,
                              void* d_out, int out_size, void* d_ws, size_t ws_size,
                              hipStream_t stream) {
  const float* vin = (const float*)d_in[0];
  float* out = (float*)d_out;
  // LDS: two padded 128x130 f32 matrices + 8192 alphas + 128 x values = 166,400 B (< 320 KB/WGP)
  const size_t shmem = (size_t)(2 * SIZE * LDA + NLAYER * NPAIR + SIZE) * sizeof(float);
  diffsort_fused_kernel<<<BATCH, 256, shmem, stream>>>(vin, out);
}